// onering_conv_layer_65326452572554
// MI455X (gfx1250) — compile-verified
//
#include <hip/hip_runtime.h>

// C[163842,64] = gather7(x)[163842,448] @ W[448,64] + b  -- fp32 WMMA 16x16x4
typedef float v2f __attribute__((ext_vector_type(2)));
typedef float v8f __attribute__((ext_vector_type(8)));

#define IN_F   64
#define OUT_F  64
#define NNB    7
#define PAIR_S 160   // float stride of one k-pair row in LDS (128 data + 32 pad)

__global__ __launch_bounds__(256)
void hexconv_wmma_f32(const float* __restrict__ x,
                      const int*   __restrict__ hex_in,
                      const float* __restrict__ W,
                      const float* __restrict__ bias,
                      float*       __restrict__ out,
                      int n_ver)
{
    // One neighbor slice of W, k-pair interleaved: Wlds[k2*PAIR_S + 2n + (k&1)]
    __shared__ float Wlds[32 * PAIR_S];   // 20480 B
    __shared__ float blds[OUT_F];

    const int tid   = threadIdx.x;
    const int wave  = tid >> 5;
    const int lane  = tid & 31;
    const int mlane = lane & 15;      // M row within tile / N col within tile
    const int khalf = lane >> 4;      // 0 -> K+{0,1}, 1 -> K+{2,3}

    if (tid < OUT_F) blds[tid] = bias[tid];

    // Each wave: 32 output rows (two 16x16 M-tiles) x full N=64 (four 16-wide N-tiles)
    const int row0 = (blockIdx.x * 8 + wave) * 32;
    const int rA   = row0 + mlane;
    const int rB   = row0 + 16 + mlane;
    const int rcA  = rA < n_ver ? rA : n_ver - 1;   // clamp so gather stays in-bounds
    const int rcB  = rB < n_ver ? rB : n_ver - 1;

    v8f acc[2][4];
#pragma unroll
    for (int t = 0; t < 2; ++t)
#pragma unroll
        for (int nt = 0; nt < 4; ++nt)
            acc[t][nt] = (v8f){};

    for (int j = 0; j < NNB; ++j) {
        __syncthreads();  // previous slice fully consumed
        // Stage W rows [j*64 .. j*64+63], pair-interleaved.  2048 pairs / 256 thr = 8 each.
        for (int p = tid; p < 32 * 64; p += 256) {
            const int k2 = p >> 6;          // 0..31
            const int n  = p & 63;
            const size_t g = (size_t)(j * IN_F + 2 * k2) * OUT_F + n;
            v2f w;
            w.x = W[g];
            w.y = W[g + OUT_F];
            *(v2f*)&Wlds[k2 * PAIR_S + 2 * n] = w;
        }
        __syncthreads();

        // per-lane gathered x rows for this neighbor (same row for lane L and L+16)
        const float* xrowA = x + (size_t)hex_in[rcA * NNB + j] * IN_F + khalf * 2;
        const float* xrowB = x + (size_t)hex_in[rcB * NNB + j] * IN_F + khalf * 2;

#pragma unroll 8
        for (int fb = 0; fb < IN_F; fb += 4) {
            // A fragments: lane supplies A[M=mlane][fb + 2*khalf + {0,1}]  (one b64 load)
            v2f aA, aB;
            aA.x = xrowA[fb];  aA.y = xrowA[fb + 1];
            aB.x = xrowB[fb];  aB.y = xrowB[fb + 1];

            // B pair row for this lane half: k2 = fb/2 + khalf
            const v2f* wrow = (const v2f*)&Wlds[((fb >> 1) + khalf) * PAIR_S];
#pragma unroll
            for (int nt = 0; nt < 4; ++nt) {
                const v2f bf = wrow[nt * 16 + mlane];   // one ds_load_b64
                acc[0][nt] = __builtin_amdgcn_wmma_f32_16x16x4_f32(
                    false, aA, false, bf, (short)0, acc[0][nt], false, false);
                acc[1][nt] = __builtin_amdgcn_wmma_f32_16x16x4_f32(
                    false, aB, false, bf, (short)0, acc[1][nt], false, false);
            }
        }
    }

    // C/D layout: VGPR v -> (M = v + 8*khalf, N = nt*16 + mlane)
    if (row0 + 32 <= n_ver) {               // fast path: every block but the last
#pragma unroll
        for (int t = 0; t < 2; ++t)
#pragma unroll
            for (int nt = 0; nt < 4; ++nt) {
                const int   n  = nt * 16 + mlane;
                const float bv = blds[n];
#pragma unroll
                for (int v = 0; v < 8; ++v) {
                    const int row = row0 + t * 16 + v + 8 * khalf;
                    out[(size_t)row * OUT_F + n] = acc[t][nt][v] + bv;
                }
            }
    } else {
#pragma unroll
        for (int t = 0; t < 2; ++t)
#pragma unroll
            for (int nt = 0; nt < 4; ++nt) {
                const int   n  = nt * 16 + mlane;
                const float bv = blds[n];
#pragma unroll
                for (int v = 0; v < 8; ++v) {
                    const int row = row0 + t * 16 + v + 8 * khalf;
                    if (row < n_ver)
                        out[(size_t)row * OUT_F + n] = acc[t][nt][v] + bv;
                }
            }
    }
}

extern "C" void kernel_launch(void* const* d_in, const int* in_sizes, int n_in,
                              void* d_out, int out_size, void* d_ws, size_t ws_size,
                              hipStream_t stream) {
    const float* x      = (const float*)d_in[0];
    const int*   hex_in = (const int*)  d_in[1];
    const float* W      = (const float*)d_in[2];
    const float* b      = (const float*)d_in[3];
    float*       out    = (float*)d_out;

    const int n_ver  = in_sizes[0] / IN_F;          // 163842
    const int blocks = (n_ver + 255) / 256;         // 256 rows per block (8 waves x 32 rows)
    hipLaunchKernelGGL(hexconv_wmma_f32, dim3(blocks), dim3(256), 0, stream,
                       x, hex_in, W, b, out, n_ver);
}